// WaveletPacketAutoencoder_18167711662096
// MI455X (gfx1250) — compile-verified
//
#include <hip/hip_runtime.h>
#include <stdint.h>

// ---------------- CDNA5 feature detection (guarded so host pass / older
// toolchains still compile; device pass on gfx1250 takes the fast paths) -----
#if __has_builtin(__builtin_amdgcn_wmma_f32_16x16x4_f32)
#define HAVE_WMMA 1
#else
#define HAVE_WMMA 0
#endif
#if __has_builtin(__builtin_amdgcn_global_load_async_to_lds_b128)
#define HAVE_ASYNC 1
#else
#define HAVE_ASYNC 0
#endif

typedef __attribute__((ext_vector_type(2))) float v2f;
typedef __attribute__((ext_vector_type(8))) float v8f;
typedef int v4i_vs __attribute__((vector_size(16)));   // matches builtin param

#define KTAPS     20
#define TILE_OUT  2048           // outputs per workgroup (8 waves x 256)
#define LDS_WORDS 4160           // staged input floats (2*TILE_OUT + halo, padded to x4)
#define NCHUNK    14             // 56 = 14 * 4 K-chunks per 256-output tile

__device__ __forceinline__ float fast_sigmoid(float x) {
  return 1.0f / (1.0f + __expf(-x));
}

// ---------------------------------------------------------------------------
// Analysis level: stride-2 20-tap conv of `in` with two filters:
//   low  = conv(in, g)                 -> outLow
//   hi   = threshold(conv(in, hAdj))   -> outHi,  hAdj[k] = h[19-k]*(-1)^k
// Banded-GEMM formulation per 256-output tile:
//   out[m,i] = sum_{j'=0}^{55} in[2*t0 + 32m + j' - 12] * W[j',i],
//   W[j',i]  = w[j' - 2i - 3]  (zero outside [0,20))
// => 14 chained V_WMMA_F32_16X16X4_F32 per filter.
// ---------------------------------------------------------------------------
__global__ __launch_bounds__(256) void wpae_analysis(
    const float* __restrict__ in, int inRowStride,
    float* __restrict__ outLow, int outLowStride,
    float* __restrict__ outHi,  int outHiStride,
    const float* __restrict__ kernelsG, const float* __restrict__ kernelsH,
    const float* __restrict__ thrP, const float* __restrict__ thrN,
    const float* __restrict__ alphaArr, const float* __restrict__ betaArr,
    int level, int Tin, int Tout)
{
  __shared__ __attribute__((aligned(16))) float lds[LDS_WORDS];

  const int  tid  = threadIdx.x;
  const int  row  = blockIdx.y;
  const int  T0   = blockIdx.x * TILE_OUT;
  const long gbase = 2L * T0 - 16;                 // global elem index of lds[0]
  const float* inRow = in + (long)row * inRowStride;

  const float* g = kernelsG + level * KTAPS;
  const float* h = kernelsH + level * KTAPS;

  const int lane = tid & 31;
  const int nn   = lane & 15;        // A row m  == B column n == lane&15
  const int hh   = (lane >> 4) & 1;  // lane half selects K = 2*hh + v
  const int wv   = tid >> 5;         // wave id 0..7

#if HAVE_WMMA
  // ---- per-thread B operands (registers), mirrored A layout: K = 2*hh + v --
  v2f Bg[NCHUNK], Bh[NCHUNK];
#pragma unroll
  for (int c = 0; c < NCHUNK; ++c) {
#pragma unroll
    for (int v = 0; v < 2; ++v) {
      const int jp = 4 * c + 2 * hh + v;
      const int k  = jp - 2 * nn - 3;
      float gv = 0.0f, hv = 0.0f;
      if (k >= 0 && k < KTAPS) {
        gv = g[k];
        hv = h[KTAPS - 1 - k] * ((k & 1) ? -1.0f : 1.0f);
      }
      Bg[c][v] = gv;
      Bh[c][v] = hv;
    }
  }
#endif

  // ---- stage input tile (with halo) into LDS ------------------------------
  const bool interior = (gbase >= 0) && (gbase + LDS_WORDS <= (long)Tin);
  if (interior) {
#if HAVE_ASYNC
    const char* gsrc  = (const char*)(inRow + gbase);
    char*       lbase = (char*)lds;
    for (int q = tid; q < LDS_WORDS / 4; q += 256) {
      __builtin_amdgcn_global_load_async_to_lds_b128(
          (__attribute__((address_space(1))) v4i_vs*)(gsrc + 16 * q),
          (__attribute__((address_space(3))) v4i_vs*)(lbase + 16 * q),
          0, 0);
    }
#if __has_builtin(__builtin_amdgcn_s_wait_asynccnt)
    __builtin_amdgcn_s_wait_asynccnt(0);
#else
    asm volatile("s_wait_asynccnt 0" ::: "memory");
#endif
#else
    const float4* gsrc = (const float4*)(inRow + gbase);
    float4*       ldst = (float4*)lds;
    for (int q = tid; q < LDS_WORDS / 4; q += 256) ldst[q] = gsrc[q];
#endif
  } else {
    for (int i = tid; i < LDS_WORDS; i += 256) {
      const long gidx = gbase + i;
      lds[i] = (gidx >= 0 && gidx < (long)Tin) ? inRow[gidx] : 0.0f;
    }
  }
  __syncthreads();

  // ---- thresholds (uniform) ----------------------------------------------
  const float tP = thrP[level], tN = thrN[level];
  const float al = alphaArr[level], be = betaArr[level];

  float* outLowRow = outLow + (long)row * outLowStride;
  float* outHiRow  = outHi  + (long)row * outHiStride;

#if HAVE_WMMA
  // ---- 28 chained WMMAs: 512 outputs (256 pos x 2 filters) per wave -------
  v8f accLo = {0.f, 0.f, 0.f, 0.f, 0.f, 0.f, 0.f, 0.f};
  v8f accHi = {0.f, 0.f, 0.f, 0.f, 0.f, 0.f, 0.f, 0.f};
  const int abase = 512 * wv + 32 * nn + 2 * hh + 4;  // LDS word for (m, j'=4c+2hh)
#pragma unroll
  for (int c = 0; c < NCHUNK; ++c) {
    const v2f a = *(const v2f*)&lds[abase + 4 * c];
    accLo = __builtin_amdgcn_wmma_f32_16x16x4_f32(false, a, false, Bg[c],
                                                  (short)0, accLo, false, false);
    accHi = __builtin_amdgcn_wmma_f32_16x16x4_f32(false, a, false, Bh[c],
                                                  (short)0, accHi, false, false);
  }
  // ---- D layout: vgpr r, lanes 0-15 -> M=r, lanes 16-31 -> M=r+8; N=lane&15
  const int obase = T0 + 256 * wv + 128 * hh + nn;
#pragma unroll
  for (int r = 0; r < 8; ++r) {
    const int pos = obase + 16 * r;
    const float lo = accLo[r];
    const float hi = accHi[r];
    const float th = hi * (fast_sigmoid(al * (hi - tP)) + fast_sigmoid(-be * (hi + tN)));
    outLowRow[pos] = lo;
    outHiRow[pos]  = th;
  }
#else
  // ---- scalar fallback (same math) ----------------------------------------
  float gw[KTAPS], hw[KTAPS];
#pragma unroll
  for (int k = 0; k < KTAPS; ++k) {
    gw[k] = g[k];
    hw[k] = h[KTAPS - 1 - k] * ((k & 1) ? -1.0f : 1.0f);
  }
  for (int qi = tid; qi < TILE_OUT; qi += 256) {
    const float* p = &lds[2 * qi + 7];  // in[2*(T0+qi) + k - 9] at word 2qi+k+7
    float aLo = 0.f, aHi = 0.f;
#pragma unroll
    for (int k = 0; k < KTAPS; ++k) {
      const float xv = p[k];
      aLo += xv * gw[k];
      aHi += xv * hw[k];
    }
    const float th = aHi * (fast_sigmoid(al * (aHi - tP)) + fast_sigmoid(-be * (aHi + tN)));
    outLowRow[T0 + qi] = aLo;
    outHiRow[T0 + qi]  = th;
  }
#endif
}

// ---------------------------------------------------------------------------
// Synthesis (adjoint of analysis, both filters summed since they are added):
//   recon[t] = sum_{k ≡ (t+9) mod 2, s=(t+9-k)/2 in [0,S)} low[s] * wsum[k]
//   wsum[k]  = g0[k] + h0[19-k]*(-1)^k
// ---------------------------------------------------------------------------
__global__ __launch_bounds__(256) void wpae_synth(
    const float* __restrict__ lowBase, int lowRowStride,
    const float* __restrict__ kernelsG, const float* __restrict__ kernelsH,
    float* __restrict__ recon, int reconRowStride,
    int S, int Trec)
{
  __shared__ float wsum[KTAPS];
  const int tid = threadIdx.x;
  if (tid < KTAPS) {
    wsum[tid] = kernelsG[tid] + kernelsH[KTAPS - 1 - tid] * ((tid & 1) ? -1.0f : 1.0f);
  }
  __syncthreads();

  const int t   = blockIdx.x * 256 + tid;
  const int row = blockIdx.y;
  if (t >= Trec) return;

  const float* low = lowBase + (long)row * lowRowStride;
  const int tp9 = t + 9;
  const int k0  = tp9 & 1;
  float acc = 0.0f;
#pragma unroll
  for (int j = 0; j < 10; ++j) {
    const int k = k0 + 2 * j;
    const int s = (tp9 - k) >> 1;
    if (s >= 0 && s < S) acc += low[s] * wsum[k];
  }
  recon[(long)row * reconRowStride + t] = acc;
}

__global__ __launch_bounds__(256) void wpae_zerofill(
    float* __restrict__ recon, int reconRowStride, int Tstart, int T)
{
  const int t = Tstart + blockIdx.x * 256 + threadIdx.x;
  if (t < T) recon[(long)blockIdx.y * reconRowStride + t] = 0.0f;
}

// ---------------------------------------------------------------------------
extern "C" void kernel_launch(void* const* d_in, const int* in_sizes, int n_in,
                              void* d_out, int out_size, void* d_ws, size_t ws_size,
                              hipStream_t stream)
{
  const float* x     = (const float*)d_in[0];
  const float* kG    = (const float*)d_in[1];
  const float* kH    = (const float*)d_in[2];
  const float* thrP  = (const float*)d_in[3];
  const float* thrN  = (const float*)d_in[4];
  const float* alpha = (const float*)d_in[5];
  const float* beta  = (const float*)d_in[6];

  const int B = 64;
  const int T = in_sizes[0] / B;               // 262144
  const int T2 = T / 2, T4 = T / 4, T8 = T / 8;

  float* recon = (float*)d_out;                // 64 x 262144
  float* det0  = recon + (long)B * T;          // 64 x 131072
  float* det1  = det0  + (long)B * T2;         // 64 x  65536
  float* det2  = det1  + (long)B * T4;         // 64 x  32768

  // Intermediate low-pass signals are packed into disjoint offsets of each
  // recon row; the recon region is fully rewritten by the last two kernels.
  const int L1OFF = 0;          // low1: 131072 floats/row
  const int L2OFF = T2;         // low2:  65536 floats/row
  const int L3OFF = T2 + T4;    // low3:  32768 floats/row

  const dim3 blk(256);

  // Level 0: x -> (det0, low1)
  wpae_analysis<<<dim3(T2 / TILE_OUT, B), blk, 0, stream>>>(
      x, T, recon + L1OFF, T, det0, T2,
      kG, kH, thrP, thrN, alpha, beta, 0, T, T2);

  // Level 1: low1 -> (det1, low2)
  wpae_analysis<<<dim3(T4 / TILE_OUT, B), blk, 0, stream>>>(
      recon + L1OFF, T, recon + L2OFF, T, det1, T4,
      kG, kH, thrP, thrN, alpha, beta, 1, T2, T4);

  // Level 2: low2 -> (det2, low3)
  wpae_analysis<<<dim3(T8 / TILE_OUT, B), blk, 0, stream>>>(
      recon + L2OFF, T, recon + L3OFF, T, det2, T8,
      kG, kH, thrP, thrN, alpha, beta, 2, T4, T8);

  // Synthesis: low3 -> recon[0 .. 2*S] ; then zero-fill the tail (which also
  // wipes the packed low1/low2/low3 scratch). Stream order guarantees safety.
  const int S    = T8;         // 32768
  const int Trec = 2 * S + 1;  // 65537
  wpae_synth<<<dim3((Trec + 255) / 256, B), blk, 0, stream>>>(
      recon + L3OFF, T, kG, kH, recon, T, S, Trec);
  wpae_zerofill<<<dim3((T - Trec + 255) / 256, B), blk, 0, stream>>>(
      recon, T, Trec, T);
}